// GraphSAGE_29901562315015
// MI455X (gfx1250) — compile-verified
//
#include <hip/hip_runtime.h>
#include <hip/hip_bf16.h>

typedef __attribute__((ext_vector_type(2))) float v2f;
typedef __attribute__((ext_vector_type(8))) float v8f;

// ---------------------------------------------------------------------------
// Guaranteed-native f32 atomic add (no CAS loop): CDNA5 GLOBAL_ATOMIC_ADD_F32,
// no-return form (tracked with STOREcnt), device scope so the RMW resolves at
// the L2 coherence point shared by all WGPs.
// ---------------------------------------------------------------------------
__device__ __forceinline__ void atomic_add_f32(float* p, float v) {
    asm volatile("global_atomic_add_f32 %0, %1, off scope:SCOPE_DEV"
                 :: "v"(p), "v"(v)
                 : "memory");
}

// ---------------------------------------------------------------------------
// Zero a float buffer (grid-stride).
// ---------------------------------------------------------------------------
__global__ void zero_f32_kernel(float* __restrict__ p, long n) {
    long i = (long)blockIdx.x * blockDim.x + threadIdx.x;
    long stride = (long)gridDim.x * blockDim.x;
    for (; i < n; i += stride) p[i] = 0.0f;
}

// ---------------------------------------------------------------------------
// In-degree: one thread per edge.
// ---------------------------------------------------------------------------
__global__ void degree_kernel(const int* __restrict__ dst, float* __restrict__ deg, long E) {
    long e = (long)blockIdx.x * blockDim.x + threadIdx.x;
    if (e < E) atomic_add_f32(&deg[dst[e]], 1.0f);
}

// ---------------------------------------------------------------------------
// Edge scatter: agg[dst] += h[src].  d = 256 floats per node.
// 64 threads per edge, float4 load per thread, 4 native f32 atomics out.
// Working set (h + agg ~ 102 MB) lives in the 192 MB L2, so this is
// L2-atomic-throughput bound; one-op-per-dword atomics are the key.
// ---------------------------------------------------------------------------
__global__ void scatter_kernel(const float* __restrict__ h,
                               const int* __restrict__ src,
                               const int* __restrict__ dst,
                               float* __restrict__ agg, long E) {
    long tid = (long)blockIdx.x * blockDim.x + threadIdx.x;
    long e = tid >> 6;               // 64 threads per edge
    int f = (int)(tid & 63) * 4;     // 4 floats per thread -> 256 features
    if (e >= E) return;
    int s = src[e];
    int d = dst[e];
    const float4 v = *(const float4*)(h + (long)s * 256 + f);
    float* o = agg + (long)d * 256 + f;
    atomic_add_f32(o + 0, v.x);
    atomic_add_f32(o + 1, v.y);
    atomic_add_f32(o + 2, v.z);
    atomic_add_f32(o + 3, v.w);
}

// ---------------------------------------------------------------------------
// Fused SAGE layer GEMM on the matrix pipe (exact f32 WMMA):
//   C[i,j] = act( A1[i,:]@B1[:,j] + (A2[i,:]/max(deg[i],1))@B2[:,j] + bias[j] )
// One wave per 16x16 C tile; K consumed 4 at a time with V_WMMA_F32_16X16X4_F32.
//
// ISA VGPR layouts (CDNA5 ISA):
//   A (16x4 f32):  lane L holds row L&15;  VGPR0/1 = K = 2*(L>>4), 2*(L>>4)+1
//   B (4x16 f32):  lane L holds col L&15;  VGPR0/1 = K = 2*(L>>4), 2*(L>>4)+1
//   C (16x16 f32): VGPR v holds row v + 8*(L>>4), col L&15
// ---------------------------------------------------------------------------
__global__ void sage_gemm_kernel(const float* __restrict__ A1,
                                 const float* __restrict__ A2,
                                 const float* __restrict__ deg,
                                 const float* __restrict__ B1,
                                 const float* __restrict__ B2,
                                 const float* __restrict__ bias,
                                 float* __restrict__ C,
                                 int M, int K, int N, int relu) {
    const int lane = threadIdx.x & 31;
    const int wave = threadIdx.x >> 5;
    const int rowTile = blockIdx.x;
    const int colTile = blockIdx.y * (blockDim.x >> 5) + wave;
    const int row0 = rowTile * 16;
    const int col0 = colTile * 16;

    const int mrow  = lane & 15;        // row within tile for A; col within tile for B/C
    const int khalf = (lane >> 4) * 2;  // k sub-offset for A/B fragments

    int arow = row0 + mrow;
    if (arow >= M) arow = M - 1;        // clamp loads; stores are guarded below
    const float scale = 1.0f / fmaxf(deg[arow], 1.0f);

    const float* a1p = A1 + (long)arow * K + khalf;
    const float* a2p = A2 + (long)arow * K + khalf;
    const int bcol = col0 + mrow;

    v8f c = {0.f, 0.f, 0.f, 0.f, 0.f, 0.f, 0.f, 0.f};

#pragma unroll 4
    for (int k = 0; k < K; k += 4) {
        // self path: A1 x B1
        v2f a1;
        a1.x = a1p[k];
        a1.y = a1p[k + 1];
        v2f b1v;
        b1v.x = B1[(long)(k + khalf) * N + bcol];
        b1v.y = B1[(long)(k + khalf + 1) * N + bcol];
        c = __builtin_amdgcn_wmma_f32_16x16x4_f32(false, a1, false, b1v,
                                                  (short)0, c, false, false);
        // neighbor path: (A2 / deg) x B2  -- mean folded into the A fragment
        v2f a2;
        a2.x = a2p[k] * scale;
        a2.y = a2p[k + 1] * scale;
        v2f b2v;
        b2v.x = B2[(long)(k + khalf) * N + bcol];
        b2v.y = B2[(long)(k + khalf + 1) * N + bcol];
        c = __builtin_amdgcn_wmma_f32_16x16x4_f32(false, a2, false, b2v,
                                                  (short)0, c, false, false);
    }

    const float bv = bias[bcol];
    const int crow0 = row0 + ((lane >> 4) << 3);
#pragma unroll
    for (int v = 0; v < 8; ++v) {
        int r = crow0 + v;
        if (r < M) {
            float val = c[v] + bv;
            if (relu) val = fmaxf(val, 0.0f);
            C[(long)r * N + bcol] = val;
        }
    }
}

// ---------------------------------------------------------------------------
// Host-side orchestration (all on `stream`; graph-capture safe).
// ---------------------------------------------------------------------------
extern "C" void kernel_launch(void* const* d_in, const int* in_sizes, int n_in,
                              void* d_out, int out_size, void* d_ws, size_t ws_size,
                              hipStream_t stream) {
    const float* feat = (const float*)d_in[0];
    const float* Ws1  = (const float*)d_in[1];
    const float* Wn1  = (const float*)d_in[2];
    const float* b1   = (const float*)d_in[3];
    const float* Ws2  = (const float*)d_in[4];
    const float* Wn2  = (const float*)d_in[5];
    const float* b2   = (const float*)d_in[6];
    const int* esrc   = (const int*)d_in[7];
    const int* edst   = (const int*)d_in[8];

    const int D = 256, D2 = 128;
    const int M = in_sizes[0] / D;      // 50000 nodes
    const long E = in_sizes[7];         // 800000 edges

    // workspace: deg[M] | agg[M*256] | h1[M*256]   (~103 MB)
    char* ws = (char*)d_ws;
    float* deg = (float*)ws;
    size_t off = ((size_t)M * sizeof(float) + 1023) & ~(size_t)1023;
    float* agg = (float*)(ws + off);
    off += (((size_t)M * D * sizeof(float)) + 1023) & ~(size_t)1023;
    float* h1 = (float*)(ws + off);
    float* out = (float*)d_out;

    const long aggN = (long)M * D;
    const long scatterThreads = E * 64;
    const int  scatterBlocks = (int)((scatterThreads + 255) / 256);

    // ---- layer 1 ----
    zero_f32_kernel<<<1024, 256, 0, stream>>>(deg, M);
    zero_f32_kernel<<<4096, 256, 0, stream>>>(agg, aggN);
    degree_kernel<<<(int)((E + 255) / 256), 256, 0, stream>>>(edst, deg, E);
    scatter_kernel<<<scatterBlocks, 256, 0, stream>>>(feat, esrc, edst, agg, E);

    dim3 g1((M + 15) / 16, D / (16 * 8));   // 8 waves/block, one 16x16 tile each
    sage_gemm_kernel<<<g1, 256, 0, stream>>>(feat, agg, deg, Ws1, Wn1, b1, h1,
                                             M, D, D, /*relu=*/1);

    // ---- layer 2 ----
    zero_f32_kernel<<<4096, 256, 0, stream>>>(agg, aggN);
    scatter_kernel<<<scatterBlocks, 256, 0, stream>>>(h1, esrc, edst, agg, E);

    dim3 g2((M + 15) / 16, D2 / (16 * 8));
    sage_gemm_kernel<<<g2, 256, 0, stream>>>(h1, agg, deg, Ws2, Wn2, b2, out,
                                             M, D, D2, /*relu=*/0);
}